// BoundaryLoss_29772713295911
// MI455X (gfx1250) — compile-verified
//
#include <hip/hip_runtime.h>
#include <cfloat>
#include <math.h>

typedef __attribute__((ext_vector_type(2))) float v2f;
typedef __attribute__((ext_vector_type(8))) float v8f;

#define B_  32
#define W_  256
#define N_  8192
#define CH  1024   // boundary points staged per LDS chunk

// One workgroup per batch; 256 threads = 8 wave32. Each wave owns TWO 16-row
// waypoint tiles (rows wave*16 and 128+wave*16) so 8 waves cover all W=256
// rows, and each B fragment (one ds_load_b64) feeds two WMMAs.
// Score s(w,n) = -2*w.p + |p|^2 via V_WMMA_F32_16X16X4_F32 with augmented
// K=4 vectors A=[-2x,-2y,-2z,1], B=[px,py,pz,|p|^2].
__global__ __launch_bounds__(256) void boundary_loss_main(
    const float* __restrict__ wp,   // [B,W,3] waypointslocal
    const float* __restrict__ bp,   // [B,N,3] boundarypoints
    const float* __restrict__ bn,   // [B,N,3] boundarynormals
    int*   __restrict__ outIdx,     // [B,W]  closest_point_idx
    float* __restrict__ wsSum)      // [B]    per-batch partial loss sums
{
    __shared__ float4 sPts[CH];   // {x, y, z, |p|^2} == two K-pair float2s
    __shared__ int    sIdx[W_];
    __shared__ float  sSum[8];

    const int b    = blockIdx.x;
    const int t    = threadIdx.x;
    const int lane = t & 31;
    const int wave = t >> 5;
    const int m    = lane & 15;      // row (A) / column (B) within 16
    const bool hi  = lane >= 16;     // lane half selects K={0,1} vs K={2,3}
    const int wbase0 = wave * 16;        // rows   0..127
    const int wbase1 = 128 + wave * 16;  // rows 128..255

    // ---- A fragments: row = [-2x, -2y, -2z, 1], K striped by lane half
    const float* wr0 = wp + ((size_t)b * W_ + (size_t)(wbase0 + m)) * 3;
    const float* wr1 = wp + ((size_t)b * W_ + (size_t)(wbase1 + m)) * 3;
    const float w0x = wr0[0], w0y = wr0[1], w0z = wr0[2];
    const float w1x = wr1[0], w1y = wr1[1], w1z = wr1[2];
    v2f afrag0, afrag1;
    afrag0.x = hi ? (-2.0f * w0z) : (-2.0f * w0x);  // K=2 : K=0
    afrag0.y = hi ? 1.0f          : (-2.0f * w0y);  // K=3 : K=1
    afrag1.x = hi ? (-2.0f * w1z) : (-2.0f * w1x);
    afrag1.y = hi ? 1.0f          : (-2.0f * w1y);

    // slot r: tile0 rows r / r+8 (by lane half); slot r+8: tile1 likewise
    float minv[16];
    int   mini[16];
#pragma unroll
    for (int r = 0; r < 16; ++r) { minv[r] = FLT_MAX; mini[r] = 0; }

    const float* bpb = bp + (size_t)b * N_ * 3;
    // pre-swizzled B read: pair index 2*(col) + hi  -> single ds_load_b64
    const v2f* sB = (const v2f*)sPts;
    const int bidx0 = 2 * m + (hi ? 1 : 0);

    for (int c0 = 0; c0 < N_; c0 += CH) {
        __syncthreads();   // previous chunk's compute done before overwrite
        // ---- stage CH points to LDS, precompute |p|^2 once per workgroup
#pragma unroll
        for (int k = 0; k < CH / 256; ++k) {
            const int li = t + k * 256;
            const float* p = bpb + (size_t)(c0 + li) * 3;
            const float x = p[0], y = p[1], z = p[2];
            sPts[li] = float4{x, y, z, x * x + y * y + z * z};
        }
        if (c0 + CH < N_)  // pull next chunk toward L2 (global_prefetch_b8)
            __builtin_prefetch(bpb + (size_t)(c0 + CH + t * 4) * 3, 0, 1);
        __syncthreads();

        // ---- 2 WMMAs per B fragment; unroll x4 to batch ds loads
#pragma unroll 4
        for (int j = 0; j < CH; j += 16) {
            const v2f bfrag = sB[2 * j + bidx0];   // col n = c0+j+m, this K half
            v8f acc = {0.f, 0.f, 0.f, 0.f, 0.f, 0.f, 0.f, 0.f};
            v8f d0 = __builtin_amdgcn_wmma_f32_16x16x4_f32(
                false, afrag0, false, bfrag, (short)0, acc, false, false);
            v8f d1 = __builtin_amdgcn_wmma_f32_16x16x4_f32(
                false, afrag1, false, bfrag, (short)0, acc, false, false);
            const int n = c0 + j + m;
#pragma unroll
            for (int r = 0; r < 8; ++r) {   // strict < keeps first index
                const float s0 = d0[r];
                const bool lt0 = s0 < minv[r];
                mini[r] = lt0 ? n  : mini[r];
                minv[r] = lt0 ? s0 : minv[r];
                const float s1 = d1[r];
                const bool lt1 = s1 < minv[r + 8];
                mini[r + 8] = lt1 ? n  : mini[r + 8];
                minv[r + 8] = lt1 ? s1 : minv[r + 8];
            }
        }
    }

    // ---- argmin across the 16 lanes holding each row (per lane-half)
#pragma unroll
    for (int r = 0; r < 16; ++r) {
#pragma unroll
        for (int off = 1; off < 16; off <<= 1) {
            const float ov = __shfl_xor(minv[r], off, 32);
            const int   oi = __shfl_xor(mini[r], off, 32);
            const bool take = (ov < minv[r]) || ((ov == minv[r]) && (oi < mini[r]));
            minv[r] = take ? ov : minv[r];
            mini[r] = take ? oi : mini[r];
        }
    }
    if (lane == 0) {
#pragma unroll
        for (int r = 0; r < 8; ++r) {
            sIdx[wbase0 + r] = mini[r];
            sIdx[wbase1 + r] = mini[r + 8];
        }
    }
    if (lane == 16) {
#pragma unroll
        for (int r = 0; r < 8; ++r) {
            sIdx[wbase0 + 8 + r] = mini[r];
            sIdx[wbase1 + 8 + r] = mini[r + 8];
        }
    }
    __syncthreads();

    // ---- epilogue: one waypoint per thread; dot, ExpRelu, block reduction
    const int idx = sIdx[t];
    const float* wr = wp  + ((size_t)b * W_ + t) * 3;
    const float* pr = bpb + (size_t)idx * 3;
    const float* nr = bn  + ((size_t)b * N_ + (size_t)idx) * 3;
    const float dot = (wr[0] - pr[0]) * nr[0]
                    + (wr[1] - pr[1]) * nr[1]
                    + (wr[2] - pr[2]) * nr[2];
    // ALPHA=1.0, BETA=0.5
    const float er = (dot >= 0.0f) ? (expf(dot) - 1.0f)
                                   : (expf(0.5f * dot) - 1.0f);
    outIdx[b * W_ + t] = idx;

    float v = er;
#pragma unroll
    for (int off = 16; off >= 1; off >>= 1) v += __shfl_xor(v, off, 32);
    if (lane == 0) sSum[wave] = v;
    __syncthreads();
    if (t == 0) {
        float s = 0.f;
#pragma unroll
        for (int i = 0; i < 8; ++i) s += sSum[i];
        wsSum[b] = s;   // unconditional overwrite: deterministic across replays
    }
}

// Fold 32 per-batch partials into the scalar mean loss (single wave).
__global__ __launch_bounds__(32) void boundary_loss_finalize(
    const float* __restrict__ wsSum, float* __restrict__ outLoss)
{
    const int lane = threadIdx.x;
    float v = (lane < B_) ? wsSum[lane] : 0.0f;
#pragma unroll
    for (int off = 16; off >= 1; off >>= 1) v += __shfl_xor(v, off, 32);
    if (lane == 0) *outLoss = v / (float)(B_ * W_);
}

extern "C" void kernel_launch(void* const* d_in, const int* in_sizes, int n_in,
                              void* d_out, int out_size, void* d_ws, size_t ws_size,
                              hipStream_t stream) {
    const float* wp = (const float*)d_in[0];   // waypointslocal  [32,256,3]
    const float* bp = (const float*)d_in[1];   // boundarypoints  [32,8192,3]
    const float* bn = (const float*)d_in[2];   // boundarynormals [32,8192,3]

    int*   outIdx  = (int*)d_out;               // first 8192 elems: int32 idx
    float* outLoss = (float*)d_out + B_ * W_;   // element 8192: scalar loss
    float* ws      = (float*)d_ws;              // 32 floats of scratch

    boundary_loss_main<<<B_, 256, 0, stream>>>(wp, bp, bn, outIdx, ws);
    boundary_loss_finalize<<<1, 32, 0, stream>>>(ws, outLoss);
}